// teacher_point2_61735859912966
// MI455X (gfx1250) — compile-verified
//
#include <hip/hip_runtime.h>
#include <hip/hip_bf16.h>

// ---------------- constants (match reference) ----------------
constexpr int BB = 8;          // batch
constexpr int NN = 2048;       // points
constexpr int NS = 32;         // nsample
constexpr int KREL = 12;       // geometry-aware k
constexpr float EPS_BN = 1e-5f;
constexpr int NUM_CLASSES = 18;

typedef __attribute__((ext_vector_type(2))) float v2f;
typedef __attribute__((ext_vector_type(8))) float v8f;

#if defined(__gfx1250__) && __has_builtin(__builtin_amdgcn_wmma_f32_16x16x4_f32)
#define HAVE_WMMA_F32 1
#else
#define HAVE_WMMA_F32 0
#endif

static inline int cdiv(long long a, long long b) { return (int)((a + b - 1) / b); }

// ---------------- transpose input + build l0p9 ----------------
__global__ void build_xyzT_l0p_kernel(const float* __restrict__ xyz,
                                      float* __restrict__ xyzT,
                                      float* __restrict__ l0p9) {
  int i = blockIdx.x * blockDim.x + threadIdx.x; // over B*N*3
  if (i >= BB * NN * 3) return;
  int c = i % 3;
  int bn = i / 3;
  int n = bn % NN;
  int b = bn / NN;
  float v = xyz[((size_t)b * 3 + c) * NN + n];
  xyzT[i] = v;
  float* l = l0p9 + (size_t)bn * 9;
  l[c] = v; l[3 + c] = v; l[6 + c] = v;
}

// ---------------- farthest point sampling (1 block / batch) ----------------
#define FPS_TPB 256
__global__ void fps_kernel(const float* __restrict__ pts, int Npts, int npoint,
                           int* __restrict__ out) {
  int b = blockIdx.x;
  const float* p = pts + (size_t)b * Npts * 3;
  int* o = out + (size_t)b * npoint;
  __shared__ float dist[2048];
  __shared__ float rv[FPS_TPB];
  __shared__ int   ri[FPS_TPB];
  __shared__ int   s_next;
  int t = threadIdx.x;
  float fx = p[0], fy = p[1], fz = p[2];
  for (int i = t; i < Npts; i += FPS_TPB) {
    float dx = p[i * 3] - fx, dy = p[i * 3 + 1] - fy, dz = p[i * 3 + 2] - fz;
    dist[i] = dx * dx + dy * dy + dz * dz;
  }
  if (t == 0) o[0] = 0;
  __syncthreads();
  for (int it = 1; it < npoint; ++it) {
    float bv = -1.f; int bi = 0x7fffffff;
    for (int i = t; i < Npts; i += FPS_TPB) {
      float v = dist[i];
      if (v > bv || (v == bv && i < bi)) { bv = v; bi = i; }
    }
    rv[t] = bv; ri[t] = bi;
    __syncthreads();
    for (int s = FPS_TPB / 2; s > 0; s >>= 1) {
      if (t < s) {
        float ov = rv[t + s]; int oi = ri[t + s];
        if (ov > rv[t] || (ov == rv[t] && oi < ri[t])) { rv[t] = ov; ri[t] = oi; }
      }
      __syncthreads();
    }
    if (t == 0) { s_next = ri[0]; o[it] = ri[0]; }
    __syncthreads();
    int nx = s_next;
    float nxx = p[nx * 3], nxy = p[nx * 3 + 1], nxz = p[nx * 3 + 2];
    for (int i = t; i < Npts; i += FPS_TPB) {
      float dx = p[i * 3] - nxx, dy = p[i * 3 + 1] - nxy, dz = p[i * 3 + 2] - nxz;
      float d = dx * dx + dy * dy + dz * dz;
      dist[i] = fminf(dist[i], d);
    }
    __syncthreads();
  }
}

// sorted(fps_idx): mark flags, compact in index order (already sorted)
__global__ void sort_idx_kernel(const int* __restrict__ in, int m, int Npts,
                                int* __restrict__ out) {
  int b = blockIdx.x;
  __shared__ unsigned char flag[2048];
  const int* ii = in + (size_t)b * m;
  int* oo = out + (size_t)b * m;
  for (int i = threadIdx.x; i < Npts; i += blockDim.x) flag[i] = 0;
  __syncthreads();
  for (int i = threadIdx.x; i < m; i += blockDim.x) flag[ii[i]] = 1;
  __syncthreads();
  if (threadIdx.x == 0) {
    int c = 0;
    for (int i = 0; i < Npts && c < m; ++i) if (flag[i]) oo[c++] = i;
    while (c < m) oo[c++] = Npts - 1;
  }
}

__global__ void gather3_kernel(const float* __restrict__ xyz, const int* __restrict__ idx,
                               float* __restrict__ out, int S, int Npts) {
  int i = blockIdx.x * blockDim.x + threadIdx.x; // over B*S
  if (i >= BB * S) return;
  int b = i / S;
  int id = idx[i];
  const float* src = xyz + ((size_t)b * Npts + id) * 3;
  float* dst = out + (size_t)i * 3;
  dst[0] = src[0]; dst[1] = src[1]; dst[2] = src[2];
}

// ---------------- ball query ----------------
__global__ void ball_query_kernel(const float* __restrict__ xyz, const float* __restrict__ nxyz,
                                  int* __restrict__ idx, int Npts, int S, float r2) {
  int i = blockIdx.x * blockDim.x + threadIdx.x; // over B*S
  if (i >= BB * S) return;
  int b = i / S;
  const float* q = nxyz + (size_t)i * 3;
  const float* p = xyz + (size_t)b * Npts * 3;
  int* o = idx + (size_t)i * NS;
  float qx = q[0], qy = q[1], qz = q[2];
  int cnt = 0, first = 0;
  for (int j = 0; j < Npts && cnt < NS; ++j) {
    float dx = p[j * 3] - qx, dy = p[j * 3 + 1] - qy, dz = p[j * 3 + 2] - qz;
    if (dx * dx + dy * dy + dz * dz <= r2) {
      if (cnt == 0) first = j;
      o[cnt++] = j;
    }
  }
  for (int k = cnt; k < NS; ++k) o[k] = first;
}

// -------- grouping with zero-padded channel stride Cinp (for guard-free GEMM) --------
__global__ void group_kernel(const float* __restrict__ xyz, const float* __restrict__ nxyz,
                             const float* __restrict__ points, const int* __restrict__ idx,
                             float* __restrict__ g, int Npts, int S, int Cpts, int Cinp) {
  int Cin = 3 + Cpts;
  size_t total = (size_t)BB * S * NS * Cinp;
  size_t i = (size_t)blockIdx.x * blockDim.x + threadIdx.x;
  if (i >= total) return;
  int c = (int)(i % Cinp);
  size_t row = i / Cinp;             // row = bs*NS + k
  size_t bs = row / NS;
  int b = (int)(bs / S);
  float v = 0.f;
  if (c < Cin) {
    int id = idx[row];
    if (c < 3) v = xyz[((size_t)b * Npts + id) * 3 + c] - nxyz[bs * 3 + c];
    else       v = points[((size_t)b * Npts + id) * Cpts + (c - 3)];
  }
  g[i] = v;
}

// -------- pad + transpose weights: Wp[n*Kp + k] = W[k*Nc + n], zero-filled --------
__global__ void padw_kernel(const float* __restrict__ W, const float* __restrict__ bias,
                            float* __restrict__ Wp, float* __restrict__ bp,
                            int K, int Nc, int Kp, int Ncp) {
  int i = blockIdx.x * blockDim.x + threadIdx.x;
  if (i < Ncp) bp[i] = (i < Nc) ? bias[i] : 0.f;
  int total = Kp * Ncp;
  if (i >= total) return;
  int k = i % Kp;
  int n = i / Kp;
  Wp[i] = (k < K && n < Nc) ? W[(size_t)k * Nc + n] : 0.f;
}

// ---------------- fp32 WMMA GEMM: D(MxNc) = A(MxKp) @ W + bias ----------------
// One wave per 16x32 output tile (2 accumulators share the A fragment).
// A has padded row stride Kp (mult of 4, zero-filled); Wp is transposed+padded
// (Ncp cols x Kp); M is a multiple of 16 -> the K-loop is totally guard-free.
__global__ __launch_bounds__(32) void gemm_bias_kernel(const float* __restrict__ A,
                                                       const float* __restrict__ Wp,
                                                       const float* __restrict__ bp,
                                                       float* __restrict__ D,
                                                       int M, int Kp, int Nc) {
  int tileM = blockIdx.x, tileN = blockIdx.y;
  int lane = threadIdx.x;
  int lo = lane & 15, hi = lane >> 4;
  int row = tileM * 16 + lo;
  int col0 = tileN * 32 + lo;
  int col1 = col0 + 16;
  const float* ap = A + (size_t)row * Kp + 2 * hi;    // lanes 0-15: K{0,1}; 16-31: K{2,3}
  const float* b0 = Wp + (size_t)col0 * Kp + 2 * hi;
  const float* b1 = Wp + (size_t)col1 * Kp + 2 * hi;
#if HAVE_WMMA_F32
  v8f acc0 = {}, acc1 = {};
#pragma unroll 4
  for (int k0 = 0; k0 < Kp; k0 += 4) {
    v2f av  = *(const v2f*)(ap + k0);   // global_load_b64, 8B aligned
    v2f bv0 = *(const v2f*)(b0 + k0);
    v2f bv1 = *(const v2f*)(b1 + k0);
    acc0 = __builtin_amdgcn_wmma_f32_16x16x4_f32(false, av, false, bv0, (short)0, acc0,
                                                 false, false);
    acc1 = __builtin_amdgcn_wmma_f32_16x16x4_f32(false, av, false, bv1, (short)0, acc1,
                                                 false, false);
  }
  float bval0 = bp[col0];
  float bval1 = bp[col1];
#pragma unroll
  for (int v = 0; v < 8; ++v) {
    int r = tileM * 16 + v + 8 * hi;   // C/D layout: vgpr v -> rows v / v+8
    if (col0 < Nc) D[(size_t)r * Nc + col0] = acc0[v] + bval0;
    if (col1 < Nc) D[(size_t)r * Nc + col1] = acc1[v] + bval1;
  }
#else
  float s0[8] = {}, s1[8] = {};
  for (int k = 0; k < Kp; ++k) {
#pragma unroll
    for (int v = 0; v < 8; ++v) {
      int r = tileM * 16 + v + 8 * hi;
      float a = A[(size_t)r * Kp + k];
      s0[v] = fmaf(a, Wp[(size_t)col0 * Kp + k], s0[v]);
      s1[v] = fmaf(a, Wp[(size_t)col1 * Kp + k], s1[v]);
    }
  }
#pragma unroll
  for (int v = 0; v < 8; ++v) {
    int r = tileM * 16 + v + 8 * hi;
    if (col0 < Nc) D[(size_t)r * Nc + col0] = s0[v] + bp[col0];
    if (col1 < Nc) D[(size_t)r * Nc + col1] = s1[v] + bp[col1];
  }
#endif
}

// ---------------- per-channel sum / sumsq (for batch-stat BN) ----------------
__global__ void zero_kernel(float* __restrict__ p, int n) {
  int i = blockIdx.x * blockDim.x + threadIdx.x;
  if (i < n) p[i] = 0.f;
}

__global__ void colstats_kernel(const float* __restrict__ X, size_t M, int C,
                                float* __restrict__ stats) {
  __shared__ float acc[1024]; // 2*Cmax, Cmax=512
  for (int i = threadIdx.x; i < 2 * C; i += blockDim.x) acc[i] = 0.f;
  __syncthreads();
  size_t total = M * (size_t)C;
  size_t start = (size_t)blockIdx.x * blockDim.x + threadIdx.x;
  size_t stride = (size_t)gridDim.x * blockDim.x;
  for (size_t i = start; i < total; i += stride) {
    float v = X[i];
    int c = (int)(i % C);
    atomicAdd(&acc[c], v);          // ds_add_f32
    atomicAdd(&acc[C + c], v * v);
  }
  __syncthreads();
  for (int i = threadIdx.x; i < 2 * C; i += blockDim.x) atomicAdd(&stats[i], acc[i]);
}

__global__ void bn_relu_kernel(float* __restrict__ X, size_t M, int C,
                               const float* __restrict__ stats,
                               const float* __restrict__ gamma,
                               const float* __restrict__ beta) {
  size_t total = M * (size_t)C;
  size_t i = (size_t)blockIdx.x * blockDim.x + threadIdx.x;
  if (i >= total) return;
  int c = (int)(i % C);
  float invM = 1.f / (float)M;
  float mean = stats[c] * invM;
  float var = stats[C + c] * invM - mean * mean;
  float y = gamma[c] * (X[i] - mean) * rsqrtf(var + EPS_BN) + beta[c];
  X[i] = fmaxf(y, 0.f);
}

// ---------------- max over nsample ----------------
__global__ void maxpool_kernel(const float* __restrict__ X, float* __restrict__ Y,
                               int BS, int C) {
  int i = blockIdx.x * blockDim.x + threadIdx.x; // over BS*C
  if (i >= BS * C) return;
  int bs = i / C, c = i % C;
  const float* p = X + (size_t)bs * NS * C + c;
  float m = p[0];
  for (int k = 1; k < NS; ++k) m = fmaxf(m, p[(size_t)k * C]);
  Y[(size_t)bs * C + c] = m;
}

// ---------------- 3-NN + interpolation weights ----------------
__global__ void knn3_kernel(const float* __restrict__ x1, const float* __restrict__ x2,
                            int* __restrict__ idx3, float* __restrict__ w3, int N1, int N2) {
  int i = blockIdx.x * blockDim.x + threadIdx.x; // over B*N1
  if (i >= BB * N1) return;
  int b = i / N1;
  const float* q = x1 + (size_t)i * 3;
  const float* p = x2 + (size_t)b * N2 * 3;
  float qx = q[0], qy = q[1], qz = q[2];
  float d0 = 1e30f, d1 = 1e30f, d2 = 1e30f;
  int i0 = 0, i1 = 0, i2 = 0;
  for (int j = 0; j < N2; ++j) {
    float dx = p[j * 3] - qx, dy = p[j * 3 + 1] - qy, dz = p[j * 3 + 2] - qz;
    float d = dx * dx + dy * dy + dz * dz;
    if (d < d0)      { d2 = d1; i2 = i1; d1 = d0; i1 = i0; d0 = d; i0 = j; }
    else if (d < d1) { d2 = d1; i2 = i1; d1 = d; i1 = j; }
    else if (d < d2) { d2 = d; i2 = j; }
  }
  float w0 = 1.f / (d0 + 1e-8f), w1 = 1.f / (d1 + 1e-8f), w2 = 1.f / (d2 + 1e-8f);
  float ws = w0 + w1 + w2;
  idx3[(size_t)i * 3] = i0; idx3[(size_t)i * 3 + 1] = i1; idx3[(size_t)i * 3 + 2] = i2;
  w3[(size_t)i * 3] = w0 / ws; w3[(size_t)i * 3 + 1] = w1 / ws; w3[(size_t)i * 3 + 2] = w2 / ws;
}

__global__ void interp_concat_kernel(const float* __restrict__ p1, const float* __restrict__ p2,
                                     const int* __restrict__ idx3, const float* __restrict__ w3,
                                     float* __restrict__ out, int N1, int N2, int C1, int C2,
                                     int Cp) {
  int Cc = C1 + C2;
  size_t total = (size_t)BB * N1 * Cp;
  size_t i = (size_t)blockIdx.x * blockDim.x + threadIdx.x;
  if (i >= total) return;
  int c = (int)(i % Cp);
  size_t bn = i / Cp;
  int b = (int)(bn / N1);
  float v = 0.f;
  if (c < C1) {
    v = p1[bn * C1 + c];
  } else if (c < Cc) {
    int cc = c - C1;
    const float* base = p2 + (size_t)b * N2 * C2;
    float s = 0.f;
    for (int t = 0; t < 3; ++t)
      s += base[(size_t)idx3[bn * 3 + t] * C2 + cc] * w3[bn * 3 + t];
    v = s;
  }
  out[i] = v;
}

// ---------------- geometry-aware relative features, mean over k ----------------
__global__ void geo_kernel(const float* __restrict__ xyz, const float* __restrict__ feat,
                           const int* __restrict__ fidx, float* __restrict__ out,
                           int Npts, int m, int C) {
  int i = blockIdx.x * blockDim.x + threadIdx.x; // over B*m
  if (i >= BB * m) return;
  int b = i / m;
  int fi = fidx[i];
  const float* p = xyz + (size_t)b * Npts * 3;
  float qx = p[fi * 3], qy = p[fi * 3 + 1], qz = p[fi * 3 + 2];
  float bd[KREL]; int bi[KREL];
#pragma unroll
  for (int t = 0; t < KREL; ++t) { bd[t] = 1e30f; bi[t] = 0; }
  for (int j = 0; j < Npts; ++j) {
    float dx = p[j * 3] - qx, dy = p[j * 3 + 1] - qy, dz = p[j * 3 + 2] - qz;
    float d = dx * dx + dy * dy + dz * dz;
    if (d < bd[KREL - 1]) {
      int t = KREL - 1;
      while (t > 0 && d < bd[t - 1]) { bd[t] = bd[t - 1]; bi[t] = bi[t - 1]; --t; }
      bd[t] = d; bi[t] = j;
    }
  }
  float* o = out + (size_t)i * (3 + C);
  float sx = 0.f, sy = 0.f, sz = 0.f;
#pragma unroll
  for (int t = 0; t < KREL; ++t) {
    int j = bi[t];
    sx += p[j * 3] - qx; sy += p[j * 3 + 1] - qy; sz += p[j * 3 + 2] - qz;
  }
  const float inv = 1.f / (float)KREL;
  o[0] = sx * inv; o[1] = sy * inv; o[2] = sz * inv;
  const float* fb = feat + (size_t)b * Npts * C;
  const float* fq = fb + (size_t)fi * C;
  for (int c = 0; c < C; ++c) {
    float s = 0.f;
    for (int t = 0; t < KREL; ++t) s += fb[(size_t)bi[t] * C + c];
    o[3 + c] = s * inv - fq[c];
  }
}

// ---------------- log-softmax over last dim ----------------
__global__ void logsoftmax_kernel(const float* __restrict__ H, float* __restrict__ out,
                                  int rows, int C) {
  int r = blockIdx.x * blockDim.x + threadIdx.x;
  if (r >= rows) return;
  const float* h = H + (size_t)r * C;
  float m = h[0];
  for (int c = 1; c < C; ++c) m = fmaxf(m, h[c]);
  float s = 0.f;
  for (int c = 0; c < C; ++c) s += __expf(h[c] - m);
  float ls = __logf(s);
  float* o = out + (size_t)r * C;
  for (int c = 0; c < C; ++c) o[c] = h[c] - m - ls;
}

// ---------------- (B,N,C) -> (B,C,N) ----------------
__global__ void transpose_out_kernel(const float* __restrict__ X, float* __restrict__ out,
                                     int Nn, int C) {
  size_t total = (size_t)BB * Nn * C;
  size_t i = (size_t)blockIdx.x * blockDim.x + threadIdx.x;
  if (i >= total) return;
  int n = (int)(i % Nn);
  size_t bc = i / Nn;
  int c = (int)(bc % C);
  int b = (int)(bc / C);
  out[i] = X[((size_t)b * Nn + n) * C + c];
}

// ================= host orchestration =================
struct LayerP { const float *W, *b, *g, *bt; int cin, cout; };

static void run_linear_bn_relu(const float* A, const LayerP& L, float* D,
                               float* Wpad, float* bpad, float* stats,
                               size_t M, hipStream_t st) {
  int Kp = (L.cin + 3) & ~3;
  int Ncp = ((L.cout + 31) / 32) * 32;
  padw_kernel<<<cdiv((long long)Kp * Ncp, 256), 256, 0, st>>>(L.W, L.b, Wpad, bpad,
                                                              L.cin, L.cout, Kp, Ncp);
  dim3 grid((int)(M / 16), cdiv(L.cout, 32));
  gemm_bias_kernel<<<grid, 32, 0, st>>>(A, Wpad, bpad, D, (int)M, Kp, L.cout);
  zero_kernel<<<cdiv(2 * L.cout, 256), 256, 0, st>>>(stats, 2 * L.cout);
  size_t tot = M * (size_t)L.cout;
  int nb = (int)((tot + 255) / 256); if (nb > 1024) nb = 1024;
  colstats_kernel<<<nb, 256, 0, st>>>(D, M, L.cout, stats);
  bn_relu_kernel<<<(int)((tot + 255) / 256), 256, 0, st>>>(D, M, L.cout, stats, L.g, L.bt);
}

struct Scratch {
  float *G0, *G1, *Wpad, *bpad, *stats;
  int *fpsI, *ballI, *knnI;
  float *knnW;
};

static void set_abstraction(const float* xyz, int Npts, const float* points, int Cpts,
                            int S, float radius, LayerP* L, int nl,
                            float* new_xyz, float* out_pool, Scratch& ws, hipStream_t st) {
  fps_kernel<<<BB, FPS_TPB, 0, st>>>(xyz, Npts, S, ws.fpsI);
  int bs = BB * S;
  gather3_kernel<<<cdiv(bs, 256), 256, 0, st>>>(xyz, ws.fpsI, new_xyz, S, Npts);
  ball_query_kernel<<<cdiv(bs, 256), 256, 0, st>>>(xyz, new_xyz, ws.ballI, Npts, S,
                                                   radius * radius);
  size_t rows = (size_t)bs * NS;
  int Cinp = (3 + Cpts + 3) & ~3;
  size_t tot = rows * (size_t)Cinp;
  group_kernel<<<(int)((tot + 255) / 256), 256, 0, st>>>(xyz, new_xyz, points, ws.ballI,
                                                         ws.G0, Npts, S, Cpts, Cinp);
  float* bufs[2] = {ws.G0, ws.G1};
  float* in = ws.G0; int which = 0;
  for (int l = 0; l < nl; ++l) {
    float* dst = bufs[which ^ 1];
    run_linear_bn_relu(in, L[l], dst, ws.Wpad, ws.bpad, ws.stats, rows, st);
    in = dst; which ^= 1;
  }
  maxpool_kernel<<<cdiv((long long)bs * L[nl - 1].cout, 256), 256, 0, st>>>(
      in, out_pool, bs, L[nl - 1].cout);
}

static void feature_prop(const float* x1, int N1, const float* x2, int N2,
                         const float* p1, int C1, const float* p2, int C2,
                         LayerP* L, int nl, float* outFinal, Scratch& ws, hipStream_t st) {
  int bn = BB * N1;
  knn3_kernel<<<cdiv(bn, 256), 256, 0, st>>>(x1, x2, ws.knnI, ws.knnW, N1, N2);
  int Cp = (C1 + C2 + 3) & ~3;
  size_t tot = (size_t)bn * Cp;
  interp_concat_kernel<<<(int)((tot + 255) / 256), 256, 0, st>>>(p1, p2, ws.knnI, ws.knnW,
                                                                 ws.G0, N1, N2, C1, C2, Cp);
  float* bufs[2] = {ws.G0, ws.G1};
  float* in = ws.G0; int which = 0;
  for (int l = 0; l < nl; ++l) {
    float* dst = (l == nl - 1) ? outFinal : bufs[which ^ 1];
    run_linear_bn_relu(in, L[l], dst, ws.Wpad, ws.bpad, ws.stats, (size_t)bn, st);
    in = dst; which ^= 1;
  }
}

extern "C" void kernel_launch(void* const* d_in, const int* in_sizes, int n_in,
                              void* d_out, int out_size, void* d_ws, size_t ws_size,
                              hipStream_t stream) {
  (void)in_sizes; (void)n_in; (void)out_size; (void)ws_size;
  const float* xyz_in = (const float*)d_in[0];

  // ---- parameters in setup_inputs() insertion order ----
  int pi = 1;
  auto nextp = [&]() { return (const float*)d_in[pi++]; };
  auto fill = [&](LayerP* L, int nl, const int* cins, const int* couts) {
    for (int l = 0; l < nl; ++l) {
      L[l].W = nextp(); L[l].b = nextp(); L[l].g = nextp(); L[l].bt = nextp();
      L[l].cin = cins[l]; L[l].cout = couts[l];
    }
  };
  LayerP sa1L[3], sa2L[3], sa3L[3], sa4L[3], fp4L[2], fp3L[2], fp2L[2], fp1L[3], h1L[1];
  { int ci[3] = {12, 32, 32},  co[3] = {32, 32, 64};   fill(sa1L, 3, ci, co); }
  { int ci[3] = {67, 64, 64},  co[3] = {64, 64, 128};  fill(sa2L, 3, ci, co); }
  { int ci[3] = {131, 128, 128}, co[3] = {128, 128, 256}; fill(sa3L, 3, ci, co); }
  { int ci[3] = {259, 256, 256}, co[3] = {256, 256, 512}; fill(sa4L, 3, ci, co); }
  { int ci[2] = {768, 256}, co[2] = {256, 256}; fill(fp4L, 2, ci, co); }
  { int ci[2] = {384, 256}, co[2] = {256, 256}; fill(fp3L, 2, ci, co); }
  { int ci[2] = {320, 256}, co[2] = {256, 128}; fill(fp2L, 2, ci, co); }
  { int ci[3] = {128, 128, 128}, co[3] = {128, 128, 128}; fill(fp1L, 3, ci, co); }
  { int ci[1] = {128}, co[1] = {128}; fill(h1L, 1, ci, co); }
  const float* h2W = nextp();
  const float* h2b = nextp();

  // ---- workspace layout ----
  char* wp = (char*)d_ws;
  auto alloc_f = [&](size_t n) { float* p = (float*)wp; wp += ((n * 4 + 255) / 256) * 256; return p; };
  auto alloc_i = [&](size_t n) { int* p = (int*)wp; wp += ((n * 4 + 255) / 256) * 256; return p; };

  float* xyzT = alloc_f((size_t)BB * NN * 3);
  float* l0p9 = alloc_f((size_t)BB * NN * 9);
  float* l1x = alloc_f((size_t)BB * 1024 * 3);
  float* l2x = alloc_f((size_t)BB * 256 * 3);
  float* l3x = alloc_f((size_t)BB * 64 * 3);
  float* l4x = alloc_f((size_t)BB * 16 * 3);
  float* l1p = alloc_f((size_t)BB * 1024 * 64);
  float* l2p = alloc_f((size_t)BB * 256 * 128);
  float* l3p = alloc_f((size_t)BB * 64 * 256);
  float* l4p = alloc_f((size_t)BB * 16 * 512);
  float* l3p2 = alloc_f((size_t)BB * 64 * 256);
  float* l2p2 = alloc_f((size_t)BB * 256 * 256);
  float* l1p2 = alloc_f((size_t)BB * 1024 * 128);
  float* l0p = alloc_f((size_t)BB * NN * 128);

  Scratch ws;
  ws.fpsI = alloc_i((size_t)BB * 1024);
  ws.ballI = alloc_i((size_t)BB * 1024 * NS);
  ws.knnI = alloc_i((size_t)BB * NN * 3);
  ws.knnW = alloc_f((size_t)BB * NN * 3);
  int* fpsg = alloc_i((size_t)BB * 512);
  int* fsort = alloc_i((size_t)BB * 512);
  ws.stats = alloc_f(1024);
  ws.Wpad = alloc_f(768 * 512);   // max Kp*Ncp (fp4: 768x256, sa4: 256x512)
  ws.bpad = alloc_f(1024);
  const size_t GSZ = (size_t)BB * 1024 * NS * 68; // largest padded activation (sa2 Cinp=68 fits in 64-wide sa1? take max)
  ws.G0 = alloc_f(GSZ);
  ws.G1 = alloc_f(GSZ);

  // ---- pipeline ----
  build_xyzT_l0p_kernel<<<cdiv(BB * NN * 3, 256), 256, 0, stream>>>(xyz_in, xyzT, l0p9);

  set_abstraction(xyzT, NN, l0p9, 9, 1024, 0.1f, sa1L, 3, l1x, l1p, ws, stream);
  set_abstraction(l1x, 1024, l1p, 64, 256, 0.2f, sa2L, 3, l2x, l2p, ws, stream);
  set_abstraction(l2x, 256, l2p, 128, 64, 0.4f, sa3L, 3, l3x, l3p, ws, stream);
  set_abstraction(l3x, 64, l3p, 256, 16, 0.8f, sa4L, 3, l4x, l4p, ws, stream);

  feature_prop(l3x, 64, l4x, 16, l3p, 256, l4p, 512, fp4L, 2, l3p2, ws, stream);
  feature_prop(l2x, 256, l3x, 64, l2p, 128, l3p2, 256, fp3L, 2, l2p2, ws, stream);
  feature_prop(l1x, 1024, l2x, 256, l1p, 64, l2p2, 256, fp2L, 2, l1p2, ws, stream);
  feature_prop(xyzT, NN, l1x, 1024, nullptr, 0, l1p2, 128, fp1L, 3, l0p, ws, stream);

  // ---- output offsets ----
  float* out = (float*)d_out;
  float* out_logp = out;                                        // (B,2048,18)
  float* out_pcrel = out + (size_t)BB * NN * NUM_CLASSES;       // (B,512,131)
  float* out_l0pT = out_pcrel + (size_t)BB * 512 * 131;         // (B,128,2048)

  // geometry-aware stage -> pc_rel directly
  fps_kernel<<<BB, FPS_TPB, 0, stream>>>(xyzT, NN, NN / 4, fpsg);
  sort_idx_kernel<<<BB, 256, 0, stream>>>(fpsg, NN / 4, NN, fsort);
  geo_kernel<<<cdiv(BB * (NN / 4), 128), 128, 0, stream>>>(xyzT, l0p, fsort, out_pcrel,
                                                           NN, NN / 4, 128);

  // head: relu(bn(l0p@W1+b1)) @ W2 + b2 -> log_softmax
  size_t rowsH = (size_t)BB * NN;
  run_linear_bn_relu(l0p, h1L[0], ws.G0, ws.Wpad, ws.bpad, ws.stats, rowsH, stream);
  {
    int Kp = 128, Ncp = 32; // head2: 128 -> 18
    padw_kernel<<<cdiv(Kp * Ncp, 256), 256, 0, stream>>>(h2W, h2b, ws.Wpad, ws.bpad,
                                                         128, NUM_CLASSES, Kp, Ncp);
    dim3 gh((int)(rowsH / 16), 1);
    gemm_bias_kernel<<<gh, 32, 0, stream>>>(ws.G0, ws.Wpad, ws.bpad, ws.G1, (int)rowsH,
                                            Kp, NUM_CLASSES);
  }
  logsoftmax_kernel<<<cdiv((int)rowsH, 128), 128, 0, stream>>>(ws.G1, out_logp, (int)rowsH,
                                                               NUM_CLASSES);

  // third output: transpose(l0p) -> (B,128,2048)
  transpose_out_kernel<<<cdiv((long long)BB * NN * 128, 256), 256, 0, stream>>>(l0p, out_l0pT,
                                                                                NN, 128);
}